// ContrastiveLoss_52140902973511
// MI455X (gfx1250) — compile-verified
//
#include <hip/hip_runtime.h>

// Problem constants (from reference): BATCH=4096, DIM=256, N=8192
#define B_ 4096
#define D_ 256
#define N_ 8192

typedef __attribute__((ext_vector_type(2))) float v2f;
typedef __attribute__((ext_vector_type(8))) float v8f;

__device__ __forceinline__ const float* row_ptr(const float* __restrict__ hi,
                                                const float* __restrict__ hj,
                                                int row) {
    return (row < B_) ? (hi + (size_t)row * D_) : (hj + (size_t)(row - B_) * D_);
}

// ---------------------------------------------------------------------------
// Kernel 1: a[r] = 0.5*||h_r||^2 ; rowsum[r] = 0
// One wave32 per row. 256 floats/row -> 8 per lane (two float4 loads).
// ---------------------------------------------------------------------------
__global__ __launch_bounds__(256) void norms_kernel(const float* __restrict__ hi,
                                                    const float* __restrict__ hj,
                                                    float* __restrict__ a,
                                                    float* __restrict__ rowsum) {
    const int lane = threadIdx.x & 31;
    const int wave = threadIdx.x >> 5;
    const int row  = blockIdx.x * 8 + wave;
    const float* p = row_ptr(hi, hj, row);

    float4 x0 = *(const float4*)(p + lane * 4);
    float4 x1 = *(const float4*)(p + 128 + lane * 4);
    float s = x0.x * x0.x + x0.y * x0.y + x0.z * x0.z + x0.w * x0.w
            + x1.x * x1.x + x1.y * x1.y + x1.z * x1.z + x1.w * x1.w;
    s += __shfl_xor(s, 1);
    s += __shfl_xor(s, 2);
    s += __shfl_xor(s, 4);
    s += __shfl_xor(s, 8);
    s += __shfl_xor(s, 16);
    if (lane == 0) {
        a[row]      = 0.5f * s;
        rowsum[row] = 0.0f;
    }
}

// ---------------------------------------------------------------------------
// Kernel 2: fused GEMM(h h^T) -> sim = exp(dot - a[r] - a[c]) -> rowsum += exp(sim)
// Block: 128 threads = 4 waves arranged 2x2; block tile = 128x128.
// Wave tile: 64x64 = 4x4 grid of 16x16 f32 WMMA accumulators.
// K loop: 64 steps of V_WMMA_F32_16X16X4_F32 (K=4 per step).
//
// f32 WMMA A-fragment layout (16x4): lane L holds M = L&15, K = 2*(L>>4)+{0,1}
//   -> per-lane float2 load at h[rowBase + (L&15)][k + 2*(L>>4)]
// B-fragment (4x16, B = h^T tile) has the mirrored layout -> identical
//   addressing with colBase.
// C/D layout: VGPR e: lanes 0-15 -> M=e, lanes 16-31 -> M=e+8; N = lane&15.
// ---------------------------------------------------------------------------
__global__ __launch_bounds__(128) void sim_kernel(const float* __restrict__ hi,
                                                  const float* __restrict__ hj,
                                                  const float* __restrict__ a,
                                                  float* __restrict__ rowsum,
                                                  float* __restrict__ pos) {
    const int lane  = threadIdx.x & 31;
    const int wave  = threadIdx.x >> 5;      // 0..3
    const int waveR = wave >> 1;             // 0..1
    const int waveC = wave & 1;              // 0..1

    const int rowBase = blockIdx.y * 128 + waveR * 64;
    const int colBase = blockIdx.x * 128 + waveC * 64;

    const int lrow = lane & 15;
    const int koff = (lane >> 4) * 2;        // K sub-offset for A/B fragments

    // Per-lane row pointers for the 4 A sub-tiles and 4 B sub-tiles.
    const float* pA[4];
    const float* pB[4];
#pragma unroll
    for (int i = 0; i < 4; ++i) {
        pA[i] = row_ptr(hi, hj, rowBase + i * 16 + lrow);
        pB[i] = row_ptr(hi, hj, colBase + i * 16 + lrow);
    }

    v8f acc[4][4] = {};                      // 128 accumulator VGPRs

#pragma unroll 2
    for (int k = 0; k < D_; k += 4) {
        v2f af[4], bf[4];
#pragma unroll
        for (int i = 0; i < 4; ++i) af[i] = *(const v2f*)(pA[i] + k + koff);
#pragma unroll
        for (int i = 0; i < 4; ++i) bf[i] = *(const v2f*)(pB[i] + k + koff);
#pragma unroll
        for (int ri = 0; ri < 4; ++ri)
#pragma unroll
            for (int ci = 0; ci < 4; ++ci)
                acc[ri][ci] = __builtin_amdgcn_wmma_f32_16x16x4_f32(
                    /*neg_a=*/false, af[ri],
                    /*neg_b=*/false, bf[ci],
                    /*c_mod=*/(short)0, acc[ri][ci],
                    /*reuse_a=*/false, /*reuse_b=*/false);
    }

    // Epilogue: sim = exp(dot - a[r] - a[c]); skip diagonal; grab pos entry;
    // butterfly-reduce per-row contributions across the 16-lane N-group.
    const int halfOff = (lane >> 4) * 8;     // M offset for upper half-wave
    const int ncol    = lane & 15;
#pragma unroll
    for (int ri = 0; ri < 4; ++ri) {
#pragma unroll
        for (int e = 0; e < 8; ++e) {
            const int   r       = rowBase + ri * 16 + e + halfOff;
            const float ar      = a[r];
            const int   partner = (r < B_) ? (r + B_) : (r - B_);
            float local = 0.0f;
#pragma unroll
            for (int ci = 0; ci < 4; ++ci) {
                const int   c   = colBase + ci * 16 + ncol;
                const float dot = acc[ri][ci][e];
                const float sim = __expf(dot - ar - a[c]);
                if (c == partner) pos[r] = sim;       // unique writer per row
                local += (c == r) ? 0.0f : __expf(sim);
            }
            // rows are uniform within each 16-lane group -> reduce over N
            local += __shfl_xor(local, 1);
            local += __shfl_xor(local, 2);
            local += __shfl_xor(local, 4);
            local += __shfl_xor(local, 8);
            if (ncol == 0) atomicAdd(&rowsum[r], local);
        }
    }
}

// ---------------------------------------------------------------------------
// Kernel 3: loss = sum_i (log(rowsum[i]) - pos[i]) / N
// ---------------------------------------------------------------------------
__global__ __launch_bounds__(256) void loss_kernel(const float* __restrict__ rowsum,
                                                   const float* __restrict__ pos,
                                                   float* __restrict__ out) {
    __shared__ float sh[256];
    float local = 0.0f;
    for (int i = threadIdx.x; i < N_; i += 256)
        local += __logf(rowsum[i]) - pos[i];
    sh[threadIdx.x] = local;
    __syncthreads();
    for (int s = 128; s > 0; s >>= 1) {
        if ((int)threadIdx.x < s) sh[threadIdx.x] += sh[threadIdx.x + s];
        __syncthreads();
    }
    if (threadIdx.x == 0) out[0] = sh[0] / (float)N_;
}

// ---------------------------------------------------------------------------
extern "C" void kernel_launch(void* const* d_in, const int* in_sizes, int n_in,
                              void* d_out, int out_size, void* d_ws, size_t ws_size,
                              hipStream_t stream) {
    const float* hi = (const float*)d_in[0];   // [4096, 256] f32
    const float* hj = (const float*)d_in[1];   // [4096, 256] f32

    float* a      = (float*)d_ws;              // N floats: 0.5*||h||^2
    float* rowsum = a + N_;                    // N floats: sum_{j!=i} exp(sim)
    float* pos    = rowsum + N_;               // N floats: sim[i, partner(i)]

    norms_kernel<<<N_ / 8, 256, 0, stream>>>(hi, hj, a, rowsum);
    sim_kernel<<<dim3(N_ / 128, N_ / 128), 128, 0, stream>>>(hi, hj, a, rowsum, pos);
    loss_kernel<<<1, 256, 0, stream>>>(rowsum, pos, (float*)d_out);
}